// EMGDCapsuleRouting_68719476789
// MI455X (gfx1250) — compile-verified
//
#include <hip/hip_runtime.h>
#include <hip/hip_bf16.h>
#include <math.h>

// MI455X / gfx1250, wave32. fp32 WMMA path: V_WMMA_F32_16X16X4_F32.
typedef __attribute__((ext_vector_type(2))) float v2f;
typedef __attribute__((ext_vector_type(8))) float v8f;

#define B_ROWS 16384
#define D_DIM  1024
#define K_DIM  512
#define ITERS  5

#define MT 64              // rows per routing block
#define KP (K_DIM + 4)     // padded LDS row stride: 516 % 64 == 4 -> conflict-free b64
#define DP (D_DIM + 4)     // padded x-tile stride: 1028 % 64 == 4

// ---------------------------------------------------------------------------
// Kernel 1: A_diff = softmax_row( (A_raw + A_raw^T) * 0.5 ), K=512
// ---------------------------------------------------------------------------
__global__ void __launch_bounds__(256)
adiff_kernel(const float* __restrict__ A_raw, float* __restrict__ A_diff) {
    const int k = blockIdx.x;     // row of A_diff
    const int t = threadIdx.x;    // 0..255, two columns each
    __shared__ float red[256];

    float v0 = 0.5f * (A_raw[k * K_DIM + t]         + A_raw[t * K_DIM + k]);
    float v1 = 0.5f * (A_raw[k * K_DIM + t + 256]   + A_raw[(t + 256) * K_DIM + k]);

    red[t] = fmaxf(v0, v1);
    __syncthreads();
    for (int s = 128; s > 0; s >>= 1) {
        if (t < s) red[t] = fmaxf(red[t], red[t + s]);
        __syncthreads();
    }
    const float m = red[0];
    __syncthreads();

    float e0 = expf(v0 - m), e1 = expf(v1 - m);
    red[t] = e0 + e1;
    __syncthreads();
    for (int s = 128; s > 0; s >>= 1) {
        if (t < s) red[t] += red[t + s];
        __syncthreads();
    }
    const float inv = 1.0f / red[0];
    A_diff[k * K_DIM + t]       = e0 * inv;
    A_diff[k * K_DIM + t + 256] = e1 * inv;
}

// ---------------------------------------------------------------------------
// Kernel 2: agreement[i] = c*s/(1+c*s)*sqrt(s),  s = || x_i @ W^T ||^2
// 16 rows per block, 4 waves, WMMA f32 16x16x4 over D=1024. u_hat never stored.
// Dynamic LDS: x tile [16][DP] + per-wave partial sums [4][16].
// ---------------------------------------------------------------------------
__global__ void __launch_bounds__(128)
agree_kernel(const float* __restrict__ x, const float* __restrict__ W,
             const float* __restrict__ log_c_p, float* __restrict__ agreement) {
    extern __shared__ float lds[];
    float* xt   = lds;                 // [16][DP]
    float* part = lds + 16 * DP;       // [4][16]

    const int row0 = blockIdx.x * 16;
    const int t    = threadIdx.x;      // 0..127

    // stage x tile (coalesced float4, padded rows stay 16B aligned: DP*4 = 4112)
    for (int i = t; i < 16 * (D_DIM / 4); i += 128) {
        const int r = i >> 8, c = i & 255;
        ((float4*)(xt + r * DP))[c] = ((const float4*)(x + (size_t)(row0 + r) * D_DIM))[c];
    }
    __syncthreads();

    const int wave = t >> 5;
    const int lane = t & 31;
    const int n    = lane & 15;           // A row / B col within tile
    const int k0   = (lane >> 4) << 1;    // K-pair select

    float rsq[8];
#pragma unroll
    for (int j = 0; j < 8; ++j) rsq[j] = 0.0f;

    for (int cti = 0; cti < 8; ++cti) {            // 8 column tiles per wave
        const int   kcol = (wave * 8 + cti) * 16 + n;
        const float* wrow = W + (size_t)kcol * D_DIM;
        v8f acc = {};
        for (int d0 = 0; d0 < D_DIM; d0 += 4) {
            v2f a = *(const v2f*)(xt + n * DP + d0 + k0);     // x[M][d0+k0..+1]
            v2f b = *(const v2f*)(wrow + d0 + k0);            // W[k][d0+k0..+1]
            acc = __builtin_amdgcn_wmma_f32_16x16x4_f32(
                false, a, false, b, (short)0, acc, false, false);
        }
#pragma unroll
        for (int j = 0; j < 8; ++j) rsq[j] += acc[j] * acc[j];
    }

    // reduce the 16 lanes sharing the same output rows (xor masks keep halves apart)
#pragma unroll
    for (int j = 0; j < 8; ++j) {
        float v = rsq[j];
        v += __shfl_xor(v, 1, 32);
        v += __shfl_xor(v, 2, 32);
        v += __shfl_xor(v, 4, 32);
        v += __shfl_xor(v, 8, 32);
        if (n == 0) part[wave * 16 + ((lane >> 4) << 3) + j] = v;
    }
    __syncthreads();

    if (t < 16) {
        float s = part[t] + part[16 + t] + part[32 + t] + part[48 + t];  // fixed order
        s = fmaxf(s, 1e-10f);
        const float c   = fminf(fmaxf(expf(log_c_p[0]), 0.01f), 10.0f);
        const float nrm = sqrtf(s);
        agreement[row0 + t] = (c * s / (1.0f + c * s)) * nrm;
    }
}

// ---------------------------------------------------------------------------
// Kernel 3: 5 routing iterations fully fused per 64-row tile. b lives in LDS.
// GEMM b@A_diff^T via WMMA f32, B-fragment reused across 4 row-subtiles.
// ---------------------------------------------------------------------------
__global__ void __launch_bounds__(256)
routing_kernel(const float* __restrict__ A_diff, const float* __restrict__ agreement,
               const float* __restrict__ b_running,
               const float* __restrict__ log_T_p, const int* __restrict__ temp_p,
               float* __restrict__ theta) {
    extern __shared__ float lds[];
    float* bmat = lds;                  // [MT][KP]
    float* b2   = lds + MT * KP;        // [MT][KP]
    float* agg  = lds + 2 * MT * KP;    // [MT]

    const int t    = threadIdx.x;       // 0..255, 8 waves
    const int row0 = blockIdx.x * MT;

    const float T_eff = fminf(fmaxf(expf(log_T_p[0]) * (float)temp_p[0], 0.05f), 10.0f);
    const float invT  = 1.0f / T_eff;

    for (int i = t; i < MT * K_DIM; i += 256) {
        const int r = i >> 9, c = i & 511;
        bmat[r * KP + c] = b_running[c];
    }
    if (t < MT) agg[t] = agreement[row0 + t];
    __syncthreads();

    const int wave = t >> 5, lane = t & 31;
    const int n    = lane & 15;
    const int k0   = (lane >> 4) << 1;
    const int grp  = t >> 2;            // softmax row 0..63
    const int gl   = t & 3;             // 4 lanes per row (shfl group)
    const int c0   = gl * 128;

    for (int iter = 0; iter < ITERS; ++iter) {
        // ---- b += agreement * softmax(b / T)  (row grp, 128 elems per lane)
        {
            float* brow = bmat + grp * KP;
            const float a_i = agg[grp];
            float m = -1e30f;
            for (int c = c0; c < c0 + 128; ++c) m = fmaxf(m, brow[c]);
            m = fmaxf(m, __shfl_xor(m, 1, 32));
            m = fmaxf(m, __shfl_xor(m, 2, 32));
            float s = 0.0f;
            for (int c = c0; c < c0 + 128; ++c) s += expf((brow[c] - m) * invT);
            s += __shfl_xor(s, 1, 32);
            s += __shfl_xor(s, 2, 32);
            const float sc = a_i / s;
            for (int c = c0; c < c0 + 128; ++c)
                brow[c] += sc * expf((brow[c] - m) * invT);
        }
        __syncthreads();

        // ---- b2 = bmat @ A_diff^T : wave owns 4 column tiles, 4 row-subtiles each
        for (int cti = 0; cti < 4; ++cti) {
            const int ct = wave * 4 + cti;
            const float* arow = A_diff + (size_t)(ct * 16 + n) * K_DIM;  // B[j][n]=A_diff[k][j]
            v8f acc0 = {}, acc1 = {}, acc2 = {}, acc3 = {};
            for (int kk = 0; kk < K_DIM; kk += 4) {
                v2f bf = *(const v2f*)(arow + kk + k0);                  // reused x4
                v2f a0 = *(const v2f*)(bmat + (0 * 16 + n) * KP + kk + k0);
                v2f a1 = *(const v2f*)(bmat + (1 * 16 + n) * KP + kk + k0);
                v2f a2 = *(const v2f*)(bmat + (2 * 16 + n) * KP + kk + k0);
                v2f a3 = *(const v2f*)(bmat + (3 * 16 + n) * KP + kk + k0);
                acc0 = __builtin_amdgcn_wmma_f32_16x16x4_f32(false, a0, false, bf, (short)0, acc0, false, false);
                acc1 = __builtin_amdgcn_wmma_f32_16x16x4_f32(false, a1, false, bf, (short)0, acc1, false, false);
                acc2 = __builtin_amdgcn_wmma_f32_16x16x4_f32(false, a2, false, bf, (short)0, acc2, false, false);
                acc3 = __builtin_amdgcn_wmma_f32_16x16x4_f32(false, a3, false, bf, (short)0, acc3, false, false);
            }
            const int colg  = ct * 16 + n;
            const int rbase = (lane >> 4) * 8;
#pragma unroll
            for (int j = 0; j < 8; ++j) {     // C/D layout: row = j + 8*(lane/16)
                b2[(0 * 16 + rbase + j) * KP + colg] = acc0[j];
                b2[(1 * 16 + rbase + j) * KP + colg] = acc1[j];
                b2[(2 * 16 + rbase + j) * KP + colg] = acc2[j];
                b2[(3 * 16 + rbase + j) * KP + colg] = acc3[j];
            }
        }
        __syncthreads();

        // ---- b += 0.1 * (b @ A_diff^T)
        for (int i = t; i < MT * K_DIM; i += 256) {
            const int r = i >> 9, c = i & 511;
            bmat[r * KP + c] += 0.1f * b2[r * KP + c];
        }
        __syncthreads();
    }

    // ---- theta = softmax(b / T) -> global
    {
        float* brow = bmat + grp * KP;
        float m = -1e30f;
        for (int c = c0; c < c0 + 128; ++c) m = fmaxf(m, brow[c]);
        m = fmaxf(m, __shfl_xor(m, 1, 32));
        m = fmaxf(m, __shfl_xor(m, 2, 32));
        float s = 0.0f;
        for (int c = c0; c < c0 + 128; ++c) s += expf((brow[c] - m) * invT);
        s += __shfl_xor(s, 1, 32);
        s += __shfl_xor(s, 2, 32);
        const float inv = 1.0f / s;
        float* orow = theta + (size_t)(row0 + grp) * K_DIM;
        for (int c = c0; c < c0 + 128; ++c)
            orow[c] = expf((brow[c] - m) * invT) * inv;
    }
}

// ---------------------------------------------------------------------------
extern "C" void kernel_launch(void* const* d_in, const int* in_sizes, int n_in,
                              void* d_out, int out_size, void* d_ws, size_t ws_size,
                              hipStream_t stream) {
    const float* x      = (const float*)d_in[0];
    const float* W      = (const float*)d_in[1];
    const float* A_raw  = (const float*)d_in[2];
    const float* log_T  = (const float*)d_in[3];
    const float* log_c  = (const float*)d_in[4];
    const float* b_run  = (const float*)d_in[5];
    const int*   temp   = (const int*)d_in[6];
    float* theta = (float*)d_out;

    float* A_diff    = (float*)d_ws;                 // 512*512 f32 = 1 MB
    float* agreement = A_diff + K_DIM * K_DIM;       // 16384 f32

    adiff_kernel<<<K_DIM, 256, 0, stream>>>(A_raw, A_diff);

    const size_t shmem_agree = (size_t)(16 * DP + 64) * sizeof(float);      // ~66 KB
    agree_kernel<<<B_ROWS / 16, 128, shmem_agree, stream>>>(x, W, log_c, agreement);

    const size_t shmem_rt = (size_t)(2 * MT * KP + MT) * sizeof(float);     // ~264 KB of 320 KB
    routing_kernel<<<B_ROWS / MT, 256, shmem_rt, stream>>>(A_diff, agreement, b_run,
                                                           log_T, temp, theta);
}